// RetentiveAttention_14851996909839
// MI455X (gfx1250) — compile-verified
//
#include <hip/hip_runtime.h>
#include <math.h>

// Problem constants (match reference)
#define Bc   16
#define Nc   10000
#define Cc   256
#define OUTc 256
#define KDc  16
#define Hc   4
#define DVc  64
#define KQD  64      // H*KD
#define ROWW 1024    // B*H*KD  (flat row width for sparse stage)

typedef __attribute__((ext_vector_type(2))) float v2f;
typedef __attribute__((ext_vector_type(8))) float v8f;

__device__ __forceinline__ float elu1(float x) {
    // elu(x) + 1  ==  x>0 ? x+1 : exp(x)
    return x > 0.0f ? x + 1.0f : __expf(x);
}

__device__ __forceinline__ v8f wmma4(v2f a, v2f b, v8f c) {
    return __builtin_amdgcn_wmma_f32_16x16x4_f32(
        /*neg_a=*/false, a, /*neg_b=*/false, b,
        /*c_mod=*/(short)0, c, /*reuse_a=*/false, /*reuse_b=*/false);
}

// ---------------------------------------------------------------------------
// Fused K+Q GEMM: one wave owns a 16-row M tile and the FULL 64 columns of
// both Q and K (8 accumulators) -> each A fragment feeds 8 WMMAs.
//   q = elu(X@Wq)+1  stored [M,64]
//   k = elu(X@Wk)+1  stored permuted kbuf[n*1024 + b*64 + col]
// A frag (16x4 f32, 2 VGPR): lanes 0-15 rows, {K=k0,k0+1}; lanes 16-31 {k0+2,k0+3}
// B frag (4x16 f32, 2 VGPR): lanes 0-15 cols @K=k0/k0+1; lanes 16-31 @K=k0+2/k0+3
// D (16x16 f32, 8 VGPR): VGPR r -> M = r + 8*(lane>>4), N = lane&15
// ---------------------------------------------------------------------------
__global__ __launch_bounds__(256) void gemm_kq_kernel(const float* __restrict__ X,
                                                      const float* __restrict__ Wq,
                                                      const float* __restrict__ Wk,
                                                      float* __restrict__ qbuf,
                                                      float* __restrict__ kbuf) {
    const int lane   = threadIdx.x & 31;
    const int wave   = threadIdx.x >> 5;
    const int tile_m = blockIdx.x * 8 + wave;   // one 16-row tile per wave
    const int l15    = lane & 15;
    const int hi     = lane >> 4;

    const float* Arow = X  + (size_t)(tile_m * 16 + l15) * Cc + 2 * hi;
    const float* Bq   = Wq + (size_t)(2 * hi) * KQD + l15;
    const float* Bk   = Wk + (size_t)(2 * hi) * KQD + l15;

    v8f aq[4] = {{0}};
    v8f ak[4] = {{0}};
#pragma unroll 2
    for (int k0 = 0; k0 < Cc; k0 += 4) {
        float2 a2 = *(const float2*)(Arow + k0);
        v2f a;
        a.x = a2.x;
        a.y = a2.y;
        const float* bq0 = Bq + (size_t)k0 * KQD;
        const float* bk0 = Bk + (size_t)k0 * KQD;
#pragma unroll
        for (int j = 0; j < 4; ++j) {
            v2f b;
            b.x = bq0[j * 16];
            b.y = bq0[j * 16 + KQD];
            aq[j] = wmma4(a, b, aq[j]);
        }
#pragma unroll
        for (int j = 0; j < 4; ++j) {
            v2f b;
            b.x = bk0[j * 16];
            b.y = bk0[j * 16 + KQD];
            ak[j] = wmma4(a, b, ak[j]);
        }
    }

#pragma unroll
    for (int j = 0; j < 4; ++j) {
        const int col = j * 16 + l15;
#pragma unroll
        for (int r = 0; r < 8; ++r) {
            const int m = tile_m * 16 + r + 8 * hi;
            qbuf[(size_t)m * KQD + col] = elu1(aq[j][r]);
            const int b = m / Nc;           // m = b*N + n
            const int n = m - b * Nc;
            kbuf[(size_t)n * ROWW + b * KQD + col] = elu1(ak[j][r]);
        }
    }
}

// ---------------------------------------------------------------------------
// V GEMM: one wave owns 16 rows x 64 cols (4 accumulators) -> each A fragment
// feeds 4 WMMAs; x re-read across column groups drops to 4x (L2-resident).
// ---------------------------------------------------------------------------
__global__ __launch_bounds__(256) void gemm_v_kernel(const float* __restrict__ X,
                                                     const float* __restrict__ Wv,
                                                     float* __restrict__ vbuf) {
    const int lane   = threadIdx.x & 31;
    const int wave   = threadIdx.x >> 5;
    const int wt     = blockIdx.x * 8 + wave;
    const int tile_m = wt >> 2;                 // 4 column groups of width 64
    const int ng     = wt & 3;
    const int l15    = lane & 15;
    const int hi     = lane >> 4;

    const float* Arow = X  + (size_t)(tile_m * 16 + l15) * Cc + 2 * hi;
    const float* Bv   = Wv + (size_t)(2 * hi) * OUTc + ng * 64 + l15;

    v8f acc[4] = {{0}};
#pragma unroll 2
    for (int k0 = 0; k0 < Cc; k0 += 4) {
        float2 a2 = *(const float2*)(Arow + k0);
        v2f a;
        a.x = a2.x;
        a.y = a2.y;
        const float* b0 = Bv + (size_t)k0 * OUTc;
#pragma unroll
        for (int j = 0; j < 4; ++j) {
            v2f b;
            b.x = b0[j * 16];
            b.y = b0[j * 16 + OUTc];
            acc[j] = wmma4(a, b, acc[j]);
        }
    }

#pragma unroll
    for (int j = 0; j < 4; ++j) {
        const int col = ng * 64 + j * 16 + l15;
#pragma unroll
        for (int r = 0; r < 8; ++r) {
            const int m = tile_m * 16 + r + 8 * hi;
            vbuf[(size_t)m * OUTc + col] = acc[j][r];
        }
    }
}

// ---------------------------------------------------------------------------
// SpMV row-broadcast: for edge e, out[rows[e]] += vals[e]*decay[h]*in[cols[e]]
// One 256-thread block per edge; each thread moves a float4 (coalesced 4KB row).
// State ([N,1024] = 41 MB) is L2-resident on the 192 MB L2, so the random
// row gathers and scatter-atomics stay on-chip.
// ---------------------------------------------------------------------------
__global__ __launch_bounds__(256) void spmv_kernel(const float* __restrict__ in,
                                                   float* __restrict__ out,
                                                   const int* __restrict__ rows,
                                                   const int* __restrict__ cols,
                                                   const float* __restrict__ vals) {
    const int e  = blockIdx.x;
    const int r  = rows[e];
    const int cl = cols[e];
    const int j0 = threadIdx.x * 4;               // 256 threads * 4 = 1024
    const int h  = (j0 & 63) >> 4;
    const float s = vals[e] * (0.1f + (float)h * (0.4f / 3.0f));

    float4 x = *(const float4*)(in + (size_t)cl * ROWW + j0);
    float* o = out + (size_t)r * ROWW + j0;
    atomicAdd(o + 0, s * x.x);
    atomicAdd(o + 1, s * x.y);
    atomicAdd(o + 2, s * x.z);
    atomicAdd(o + 3, s * x.w);
}

// acc += a + b  (float4 granularity)
__global__ __launch_bounds__(256) void add2_kernel(float4* __restrict__ acc,
                                                   const float4* __restrict__ a,
                                                   const float4* __restrict__ b,
                                                   size_t n4) {
    size_t i = (size_t)blockIdx.x * blockDim.x + threadIdx.x;
    if (i < n4) {
        float4 A = acc[i];
        float4 p = a[i];
        float4 r = b[i];
        A.x += p.x + r.x;
        A.y += p.y + r.y;
        A.z += p.z + r.z;
        A.w += p.w + r.w;
        acc[i] = A;
    }
}

// ---------------------------------------------------------------------------
// Finalize: per (b,n) row — w[h] = mean_d(kacc*q), v *= w[h], per-head
// LayerNorm over DV=64, permuted store (t=b*N+n  ->  out[t%B][t/B][:]).
// ---------------------------------------------------------------------------
__global__ __launch_bounds__(256) void finalize_kernel(
    const float* __restrict__ kacc,   // [N,1024]: n*1024 + b*64 + h*16 + d
    const float* __restrict__ q,      // [B*N, 64]
    const float* __restrict__ v,      // [B*N, 256]
    const float* __restrict__ gamma,  // [64]
    const float* __restrict__ beta,   // [64]
    float* __restrict__ out) {
    __shared__ float s_p[KQD];
    __shared__ float s_w[Hc];
    __shared__ float s_sum[256];
    __shared__ float s_sq[256];

    const int t = blockIdx.x;         // t = b*N + n
    const int b = t / Nc;
    const int n = t - b * Nc;
    const int c = threadIdx.x;        // 0..255
    const int h = c >> 6;
    const int dv = c & 63;

    if (c < KQD) {
        float kk = kacc[(size_t)n * ROWW + b * KQD + c];
        float qq = q[(size_t)t * KQD + c];
        s_p[c] = kk * qq;
    }
    __syncthreads();
    if (c < Hc) {
        float s = 0.0f;
#pragma unroll
        for (int d = 0; d < KDc; ++d) s += s_p[c * KDc + d];
        s_w[c] = s * (1.0f / (float)KDc);
    }
    __syncthreads();

    float vv = v[(size_t)t * OUTc + c] * s_w[h];
    s_sum[c] = vv;
    s_sq[c]  = vv * vv;
    __syncthreads();
#pragma unroll
    for (int off = 32; off >= 1; off >>= 1) {
        if (dv < off) {
            s_sum[c] += s_sum[c + off];
            s_sq[c]  += s_sq[c + off];
        }
        __syncthreads();
    }
    const float mu  = s_sum[h * DVc] * (1.0f / (float)DVc);
    const float var = s_sq[h * DVc] * (1.0f / (float)DVc) - mu * mu;
    const float y = (vv - mu) * rsqrtf(var + 1e-5f) * gamma[dv] + beta[dv];

    const long dst = (long)(t % Bc) * ((long)Nc * OUTc) + (long)(t / Bc) * OUTc + c;
    out[dst] = y;
}

// ---------------------------------------------------------------------------
extern "C" void kernel_launch(void* const* d_in, const int* in_sizes, int n_in,
                              void* d_out, int out_size, void* d_ws, size_t ws_size,
                              hipStream_t stream) {
    const float* x     = (const float*)d_in[0];
    const float* evals = (const float*)d_in[1];
    const float* Wk    = (const float*)d_in[2];
    const float* Wq    = (const float*)d_in[3];
    const float* Wv    = (const float*)d_in[4];
    const float* gamma = (const float*)d_in[5];
    const float* beta  = (const float*)d_in[6];
    const int*   erows = (const int*)d_in[7];
    const int*   ecols = (const int*)d_in[8];
    const int E = in_sizes[1];

    const size_t M = (size_t)Bc * Nc;               // 160000 rows
    const size_t flatElems = (size_t)Nc * ROWW;     // 10,240,000

    char* ws = (char*)d_ws;
    float* qbuf = (float*)ws;  ws += M * KQD * sizeof(float);
    float* kbuf = (float*)ws;  ws += flatElems * sizeof(float);   // k, then acc (in place)
    float* it1  = (float*)ws;  ws += flatElems * sizeof(float);
    float* it2  = (float*)ws;  ws += flatElems * sizeof(float);
    float* vbuf = (float*)ws;  ws += M * OUTc * sizeof(float);

    // --- GEMMs (WMMA f32 16x16x4) ---
    {
        const int mtiles = (int)(M / 16);                     // 10000
        gemm_kq_kernel<<<mtiles / 8, 256, 0, stream>>>(x, Wq, Wk, qbuf, kbuf);
        const int wtiles = mtiles * 4;                        // 40000
        gemm_v_kernel<<<wtiles / 8, 256, 0, stream>>>(x, Wv, vbuf);
    }

    // --- sparse propagation, 2 rounds (L2-resident state, atomic scatter) ---
    hipMemsetAsync(it1, 0, flatElems * sizeof(float), stream);
    spmv_kernel<<<E, 256, 0, stream>>>(kbuf, it1, erows, ecols, evals);
    hipMemsetAsync(it2, 0, flatElems * sizeof(float), stream);
    spmv_kernel<<<E, 256, 0, stream>>>(it1, it2, erows, ecols, evals);

    // acc = k + it1 + it2 (in place into kbuf)
    {
        const size_t n4 = flatElems / 4;
        const int blocks = (int)((n4 + 255) / 256);
        add2_kernel<<<blocks, 256, 0, stream>>>((float4*)kbuf, (const float4*)it1,
                                                (const float4*)it2, n4);
    }

    // --- fused epilogue: head weights, scale, per-head LayerNorm, permuted store ---
    finalize_kernel<<<(int)M, 256, 0, stream>>>(kbuf, qbuf, vbuf, gamma, beta,
                                                (float*)d_out);
}